// SearchNet_3917010173953
// MI455X (gfx1250) — compile-verified
//
#include <hip/hip_runtime.h>
#include <hip/hip_bf16.h>
#include <math.h>

// Cosine-similarity search: q[512] vs db[500000][512], top-k values+indices.
// Bandwidth-bound: 1.02 GB streamed once (~44us at 23.3 TB/s). Dot products
// run on v_wmma_f32_16x16x4_f32 (16-row tiles per wave, query replicated
// across B columns); row norms on VALU; device-side radix-select top-k.

#define EPSV 1e-8f
#define DIMS 512
#define BINS 4096
#define CAP  4096

typedef float v2f __attribute__((ext_vector_type(2)));
typedef float v8f __attribute__((ext_vector_type(8)));

__device__ __forceinline__ float bperm_f(int srcLane, float v) {
  return __int_as_float(__builtin_amdgcn_ds_bpermute(srcLane << 2, __float_as_int(v)));
}

// order-preserving map: larger float -> larger unsigned
__device__ __forceinline__ unsigned orderKey(unsigned b) {
  return b ^ ((b & 0x80000000u) ? 0xFFFFFFFFu : 0x80000000u);
}

// ---------------------------------------------------------------------------
// Kernel A: sims[r] = dot(db[r], q) / (max(||db[r]||,eps) * max(||q||,eps))
// One 16-row tile per wave. Lane l and lane l+16 own row (l&15) with disjoint
// K-halves, matching the WMMA f32 16x16x4 A layout (lanes 0-15: logical K{0,1},
// lanes 16-31: logical K{2,3}). B holds the matching query chunk replicated
// across all 16 columns, so every column of D equals dot(row, q).
// ---------------------------------------------------------------------------
__global__ void __launch_bounds__(256)
sims_kernel(const float* __restrict__ q, const float* __restrict__ db,
            float* __restrict__ sims, int n) {
  __shared__ __align__(16) float qs[DIMS];
  for (int t = threadIdx.x; t < DIMS; t += blockDim.x) qs[t] = q[t];
  __syncthreads();

  const int lane = (int)(threadIdx.x & 31u);
  const int wave = threadIdx.x >> 5;
  const int tile = blockIdx.x * 8 + wave;
  const int numTiles = (n + 15) >> 4;
  if (tile >= numTiles) return;  // uniform per wave -> EXEC stays all-ones

  // ||q||^2 : each wave computes redundantly from LDS (negligible cost)
  float qn2 = 0.f;
#pragma unroll
  for (int j = 0; j < DIMS / 32; ++j) {
    float v = qs[lane + 32 * j];
    qn2 = fmaf(v, v, qn2);
  }
#pragma unroll
  for (int m = 1; m < 32; m <<= 1) qn2 += bperm_f(lane ^ m, qn2);
  const float qn = sqrtf(qn2);

  const int base = tile * 16;
  int row = base + (lane & 15);
  if (row > n - 1) row = n - 1;                 // clamp tail rows (stores masked)
  const int half4 = (lane >> 4) << 2;           // 0 for lanes 0-15, 4 for 16-31
  const float* rp = db + (size_t)row * DIMS + half4;

  v8f c = {0.f, 0.f, 0.f, 0.f, 0.f, 0.f, 0.f, 0.f};
  float nrm = 0.f;

#pragma unroll 4
  for (int i = 0; i < DIMS / 8; ++i) {
    const float4 a  = *(const float4*)(rp + i * 8);          // db row chunk
    const float4 qv = *(const float4*)(qs + i * 8 + half4);  // matching q chunk
    nrm = fmaf(a.x, a.x, nrm);
    nrm = fmaf(a.y, a.y, nrm);
    nrm = fmaf(a.z, a.z, nrm);
    nrm = fmaf(a.w, a.w, nrm);
    v2f A0 = {a.x, a.y}, B0 = {qv.x, qv.y};
    v2f A1 = {a.z, a.w}, B1 = {qv.z, qv.w};
    c = __builtin_amdgcn_wmma_f32_16x16x4_f32(false, A0, false, B0,
                                              (short)0, c, false, false);
    c = __builtin_amdgcn_wmma_f32_16x16x4_f32(false, A1, false, B1,
                                              (short)0, c, false, false);
  }

  // full row norm: combine the two K-halves (lane l <-> lane l^16)
  nrm += bperm_f(lane ^ 16, nrm);

  // Extract dots from C layout: VGPR j, lanes 0-15 -> M=j ; lanes 16-31 -> M=j+8.
  // All columns identical, so val(lane) = c[lane&7] gives:
  //   lanes 0-7: dot(base+lane), lanes 16-23: dot(base+lane-8), ...
  float val = c[0];
#pragma unroll
  for (int j = 1; j < 8; ++j) val = ((lane & 7) == j) ? c[j] : val;
  const int src = (lane & 8) ? (lane + 8) : lane;  // valid for lanes 0..15
  const float dot = bperm_f(src, val);

  if (lane < 16) {
    const int r = base + lane;
    if (r < n) {
      const float den = fmaxf(sqrtf(nrm), EPSV) * fmaxf(qn, EPSV);
      sims[r] = dot / den;
    }
  }
}

// ---------------------------------------------------------------------------
// Top-k over sims[] (k read from device scalar; graph-capture safe).
// ---------------------------------------------------------------------------
__global__ void zero_kernel(unsigned* __restrict__ hist, unsigned* __restrict__ ctrl) {
  const int i = blockIdx.x * blockDim.x + threadIdx.x;
  if (i < BINS) hist[i] = 0u;
  if (i < 2) ctrl[i] = 0u;
}

__global__ void __launch_bounds__(256)
hist_kernel(const float* __restrict__ sims, int n, unsigned* __restrict__ hist) {
  __shared__ unsigned h[BINS];
  for (int i = threadIdx.x; i < BINS; i += blockDim.x) h[i] = 0u;
  __syncthreads();
  for (int i = blockIdx.x * blockDim.x + threadIdx.x; i < n;
       i += gridDim.x * blockDim.x) {
    const unsigned u = orderKey(__float_as_uint(sims[i]));
    atomicAdd(&h[u >> 20], 1u);
  }
  __syncthreads();
  for (int i = threadIdx.x; i < BINS; i += blockDim.x)
    if (h[i]) atomicAdd(&hist[i], h[i]);
}

__global__ void scan_kernel(const unsigned* __restrict__ hist,
                            const int* __restrict__ topk,
                            unsigned* __restrict__ ctrl) {
  if (threadIdx.x == 0 && blockIdx.x == 0) {
    const int k = topk[0];
    unsigned cum = 0, t = 0;
    for (int b = BINS - 1; b >= 0; --b) {
      cum += hist[b];
      if ((int)cum >= k) { t = (unsigned)b; break; }
    }
    ctrl[0] = t;   // threshold bin
    ctrl[1] = 0u;  // candidate counter
  }
}

__global__ void __launch_bounds__(256)
compact_kernel(const float* __restrict__ sims, int n,
               unsigned* __restrict__ ctrl, uint2* __restrict__ cand) {
  const unsigned t = ctrl[0];
  for (int i = blockIdx.x * blockDim.x + threadIdx.x; i < n;
       i += gridDim.x * blockDim.x) {
    const unsigned b = __float_as_uint(sims[i]);
    const unsigned u = orderKey(b);
    if ((u >> 20) >= t) {
      const unsigned pos = atomicAdd(&ctrl[1], 1u);
      if (pos < CAP) cand[pos] = make_uint2(b, (unsigned)i);
    }
  }
}

__global__ void __launch_bounds__(256)
select_kernel(const uint2* __restrict__ cand, const unsigned* __restrict__ ctrl,
              const int* __restrict__ topk, float* __restrict__ outv) {
  __shared__ uint2 s[CAP];
  const unsigned cc = ctrl[1];
  const int c = (int)(cc < (unsigned)CAP ? cc : (unsigned)CAP);
  const int k = topk[0];
  for (int i = threadIdx.x; i < c; i += blockDim.x) s[i] = cand[i];
  __syncthreads();
  for (int i = threadIdx.x; i < c; i += blockDim.x) {
    const unsigned ui = orderKey(s[i].x);
    const unsigned ii = s[i].y;
    int rank = 0;
    for (int j = 0; j < c; ++j) {
      const unsigned uj = orderKey(s[j].x);
      rank += (int)((uj > ui) || (uj == ui && s[j].y < ii));  // jax tie-break
    }
    if (rank < k) {
      outv[rank] = __uint_as_float(s[i].x);      // values[0..k)
      ((int*)outv)[k + rank] = (int)ii;          // indices[k..2k) as int32
    }
  }
}

extern "C" void kernel_launch(void* const* d_in, const int* in_sizes, int n_in,
                              void* d_out, int out_size, void* d_ws, size_t ws_size,
                              hipStream_t stream) {
  (void)n_in; (void)out_size; (void)ws_size;
  const float* q    = (const float*)d_in[0];
  const float* db   = (const float*)d_in[1];
  const int*   topk = (const int*)d_in[2];
  const int n = in_sizes[1] / DIMS;  // 500000

  // workspace layout (all 8B-aligned for n=500000)
  float*    sims = (float*)d_ws;                  // n floats (2.0 MB)
  unsigned* hist = (unsigned*)(sims + n);         // BINS u32 (16 KB)
  unsigned* ctrl = hist + BINS;                   // 2 u32 (+pad)
  uint2*    cand = (uint2*)(ctrl + 4);            // CAP uint2 (32 KB)

  const int numTiles = (n + 15) / 16;
  const int blocksA  = (numTiles + 7) / 8;        // 8 waves (tiles) per block

  hipLaunchKernelGGL(sims_kernel, dim3(blocksA), dim3(256), 0, stream,
                     q, db, sims, n);
  hipLaunchKernelGGL(zero_kernel, dim3((BINS + 255) / 256), dim3(256), 0, stream,
                     hist, ctrl);
  hipLaunchKernelGGL(hist_kernel, dim3(1024), dim3(256), 0, stream,
                     sims, n, hist);
  hipLaunchKernelGGL(scan_kernel, dim3(1), dim3(32), 0, stream,
                     hist, topk, ctrl);
  hipLaunchKernelGGL(compact_kernel, dim3(1024), dim3(256), 0, stream,
                     sims, n, ctrl, cand);
  hipLaunchKernelGGL(select_kernel, dim3(1), dim3(256), 0, stream,
                     cand, ctrl, topk, (float*)d_out);
}